// GridPooling_27874337751670
// MI455X (gfx1250) — compile-verified
//
#include <hip/hip_runtime.h>
#include <hip/hip_bf16.h>
#include <stdint.h>

// ---------------------------------------------------------------------------
// GridPooling on gfx1250 (MI455X):
//   keys -> dense 2^30 bitmap -> hierarchical popcount prefix -> cluster rank
//   proj = feat @ W + b via V_WMMA_F32_16X16X4_F32, fused atomic segment-max
//   W/bias staged to LDS with GLOBAL_LOAD_ASYNC_TO_LDS_B128 (ASYNCcnt)
//   coord segment-mean via float atomics, counts via uint atomics
// ---------------------------------------------------------------------------

typedef __attribute__((ext_vector_type(2))) float v2f;
typedef __attribute__((ext_vector_type(8))) float v8f;
typedef __attribute__((address_space(3))) float lds_float;

#define CIN   32
#define COUT  64
#define NWORDS  (1u << 25)          // 2^30 key bits / 32
#define GROUPW  8u                   // words per prefix group (256 keys)
#define NGROUPS (1u << 22)           // NWORDS / GROUPW
#define ORD_NEG_INF 0x007FFFFFu      // order-encoded -inf

// branchless order-preserving float<->uint mapping for atomic max (3 VALU each)
__device__ __forceinline__ unsigned f2ord(float f) {
    unsigned u = __float_as_uint(f);
    return u ^ ((unsigned)((int)u >> 31) | 0x80000000u);
}
__device__ __forceinline__ float ord2f(unsigned u) {
    return __uint_as_float(u ^ (~(unsigned)((int)u >> 31) | 0x80000000u));
}

// async 16-byte global->LDS copy (per enabled lane), tracked by ASYNCcnt
__device__ __forceinline__ void async_copy_b128(float* ldsDst, const float* gSrc) {
    unsigned ldsOff = (unsigned)(uintptr_t)(lds_float*)ldsDst;
    asm volatile("global_load_async_to_lds_b128 %0, %1, off"
                 :: "v"(ldsOff), "v"(gSrc) : "memory");
}
__device__ __forceinline__ void wait_asynccnt0() {
    asm volatile("s_wait_asynccnt 0x0" ::: "memory");
}

// ---------------------------------------------------------------------------
__global__ void k_fill_u32(unsigned* p, unsigned v, size_t n) {
    size_t i = (size_t)blockIdx.x * 256 + threadIdx.x;
    if (i < n) p[i] = v;
}

// keys + bitmap set
__global__ void k_keys_bits(const int* __restrict__ grid_coord,
                            const int* __restrict__ batch,
                            unsigned* __restrict__ keys,
                            unsigned* __restrict__ bitmap, int N) {
    int i = blockIdx.x * 256 + threadIdx.x;
    if (i >= N) return;
    unsigned gx = (unsigned)(grid_coord[3 * i + 0]) >> 1;
    unsigned gy = (unsigned)(grid_coord[3 * i + 1]) >> 1;
    unsigned gz = (unsigned)(grid_coord[3 * i + 2]) >> 1;
    unsigned bt = (unsigned)batch[i];
    unsigned key = (bt << 27) | (gx << 18) | (gy << 9) | gz;
    keys[i] = key;
    atomicOr(&bitmap[key >> 5], 1u << (key & 31u));
}

// per-group popcount sums (group = 8 words = 256 key slots)
__global__ void k_group_sums(const unsigned* __restrict__ bitmap,
                             unsigned* __restrict__ groupPref) {
    unsigned g = blockIdx.x * 256 + threadIdx.x;
    if (g >= NGROUPS) return;
    unsigned base = g * GROUPW, s = 0;
#pragma unroll
    for (unsigned w = 0; w < GROUPW; ++w) s += __popc(bitmap[base + w]);
    groupPref[g] = s;
}

// exclusive scan of 256-element tiles in LDS; emits per-block totals
__global__ void k_scan256(unsigned* __restrict__ data,
                          unsigned* __restrict__ blockSums, unsigned n) {
    __shared__ unsigned sh[256];
    unsigned g = blockIdx.x * 256 + threadIdx.x;
    unsigned v = (g < n) ? data[g] : 0u;
    sh[threadIdx.x] = v;
    __syncthreads();
    for (int off = 1; off < 256; off <<= 1) {
        unsigned t = (threadIdx.x >= (unsigned)off) ? sh[threadIdx.x - off] : 0u;
        __syncthreads();
        sh[threadIdx.x] += t;
        __syncthreads();
    }
    unsigned incl = sh[threadIdx.x];
    if (g < n) data[g] = incl - v;                        // exclusive
    if (threadIdx.x == 255) blockSums[blockIdx.x] = incl; // block total
}

// single-block sequential-chunk exclusive scan (for block sums)
__global__ void k_scan_single(unsigned* __restrict__ data, unsigned n) {
    __shared__ unsigned sh[256];
    __shared__ unsigned carry;
    if (threadIdx.x == 0) carry = 0;
    __syncthreads();
    for (unsigned base = 0; base < n; base += 256) {
        unsigned g = base + threadIdx.x;
        unsigned v = (g < n) ? data[g] : 0u;
        sh[threadIdx.x] = v;
        __syncthreads();
        for (int off = 1; off < 256; off <<= 1) {
            unsigned t = (threadIdx.x >= (unsigned)off) ? sh[threadIdx.x - off] : 0u;
            __syncthreads();
            sh[threadIdx.x] += t;
            __syncthreads();
        }
        unsigned incl = sh[threadIdx.x];
        if (g < n) data[g] = carry + incl - v;
        __syncthreads();
        if (threadIdx.x == 255) carry += incl;
        __syncthreads();
    }
}

__global__ void k_add_offsets(unsigned* __restrict__ data,
                              const unsigned* __restrict__ blockSums, unsigned n) {
    unsigned g = blockIdx.x * 256 + threadIdx.x;
    if (g < n) data[g] += blockSums[g >> 8];
}

// rank lookup -> cluster id; counts; coord sums; cluster output
__global__ void k_cluster(const unsigned* __restrict__ keys,
                          const unsigned* __restrict__ bitmap,
                          const unsigned* __restrict__ groupPref,
                          const float* __restrict__ coord,
                          unsigned* __restrict__ cluster,
                          unsigned* __restrict__ counts,
                          float* __restrict__ outCoordSum,
                          float* __restrict__ outCluster, int N) {
    int i = blockIdx.x * 256 + threadIdx.x;
    if (i >= N) return;
    unsigned key = keys[i];
    unsigned g = key >> 8;
    unsigned rank = groupPref[g];
    unsigned w0 = g * GROUPW;
    unsigned wi = (key >> 5) & 7u;
    for (unsigned w = 0; w < wi; ++w) rank += __popc(bitmap[w0 + w]);
    unsigned bit = key & 31u;
    rank += __popc(bitmap[w0 + wi] & ((1u << bit) - 1u));
    cluster[i] = rank;
    outCluster[i] = (float)rank;
    atomicAdd(&counts[rank], 1u);
    unsigned cb = rank * 3u;
    atomicAdd(&outCoordSum[cb + 0u], coord[3 * (size_t)i + 0]);
    atomicAdd(&outCoordSum[cb + 1u], coord[3 * (size_t)i + 1]);
    atomicAdd(&outCoordSum[cb + 2u], coord[3 * (size_t)i + 2]);
}

// fused GEMM (f32 WMMA 16x16x4) + atomic segment-max scatter.
// block = 256 threads = 8 waves; each wave: 16 points x 64 outputs.
__global__ void k_gemm_segmax(const float* __restrict__ feat,
                              const float* __restrict__ W,
                              const float* __restrict__ bias,
                              const unsigned* __restrict__ cluster,
                              unsigned* __restrict__ featMaxOrd, int N) {
    __shared__ float shW[CIN * COUT];  // 8 KB
    __shared__ float shB[COUT];        // 256 B

    // Async direct-to-LDS staging of W + bias: 512 + 16 b128 chunks over 256
    // threads, no VGPR round trip, tracked with ASYNCcnt.
    {
        unsigned t = threadIdx.x;
#pragma unroll
        for (int c = 0; c < 2; ++c) {
            unsigned chunk = t + (unsigned)c * 256u;         // 0..511
            async_copy_b128(&shW[chunk * 4u], W + chunk * 4u);
        }
        if (t < 16u) async_copy_b128(&shB[t * 4u], bias + t * 4u);
        wait_asynccnt0();
    }
    __syncthreads();

    int wave = threadIdx.x >> 5;
    int lane = threadIdx.x & 31;
    int base = (blockIdx.x * 8 + wave) * 16;   // wave-uniform
    if (base >= N) return;                     // uniform exit: EXEC stays all-1s

    int m = lane & 15;
    int h = lane >> 4;                         // 0: K pair {0,1}, 1: K pair {2,3}

    // A-matrix fragments: 16x4 f32 per k-step, 8 steps cover K=32.
    // lane<16 holds (K=4s+0, 4s+1), lane>=16 holds (K=4s+2, 4s+3) of row m.
    v2f av[8];
    const float* frow = feat + (size_t)(base + m) * CIN + 2 * h;
#pragma unroll
    for (int s = 0; s < 8; ++s)
        av[s] = *(const v2f*)(frow + 4 * s);   // 8-byte aligned

    // cluster element offsets for this wave's 16 points
    // (C layout: VGPR r -> M = r + 8*h); unsigned 32-bit offsets keep the
    // atomics in SADDR-base + voffset form (no 64-bit VALU address math).
    unsigned cl[8];
#pragma unroll
    for (int r = 0; r < 8; ++r)
        cl[r] = cluster[base + r + 8 * h] * (unsigned)COUT;

#pragma unroll
    for (int nt = 0; nt < 4; ++nt) {
        unsigned n = (unsigned)(nt * 16 + m);  // this lane's output column
        v8f c = {};
#pragma unroll
        for (int s = 0; s < 8; ++s) {
            // B 4x16 tile from LDS: lane<16 rows {4s+0,4s+1}; lane>=16 {4s+2,4s+3}
            v2f bv;
            bv.x = shW[(unsigned)(4 * s + 2 * h + 0) * COUT + n];
            bv.y = shW[(unsigned)(4 * s + 2 * h + 1) * COUT + n];
            c = __builtin_amdgcn_wmma_f32_16x16x4_f32(
                    false, av[s], false, bv, (short)0, c, false, false);
        }
        float bb = shB[n];
#pragma unroll
        for (int r = 0; r < 8; ++r) {
            unsigned ov = f2ord(c[r] + bb);
            atomicMax(&featMaxOrd[cl[r] + n], ov);
        }
    }
}

// decode ordered-uint maxima; zero empty segments (matches jnp.where mask)
__global__ void k_finalize_feat(float* __restrict__ outFeat,
                                const unsigned* __restrict__ counts, int N) {
    size_t i = (size_t)blockIdx.x * 256 + threadIdx.x;
    size_t total = (size_t)N * COUT;
    if (i >= total) return;
    unsigned s = (unsigned)(i >> 6);
    unsigned u = ((const unsigned*)outFeat)[i];
    outFeat[i] = (counts[s] > 0u) ? ord2f(u) : 0.0f;
}

__global__ void k_finalize_coord(float* __restrict__ outCoord,
                                 const unsigned* __restrict__ counts, int N) {
    size_t i = (size_t)blockIdx.x * 256 + threadIdx.x;
    size_t total = (size_t)N * 3;
    if (i >= total) return;
    unsigned s = (unsigned)(i / 3);
    float cnt = (float)counts[s];
    outCoord[i] = outCoord[i] / fmaxf(cnt, 1.0f);
}

// ---------------------------------------------------------------------------
extern "C" void kernel_launch(void* const* d_in, const int* in_sizes, int n_in,
                              void* d_out, int out_size, void* d_ws, size_t ws_size,
                              hipStream_t stream) {
    const float* feat       = (const float*)d_in[0];   // N x 32
    const float* coord      = (const float*)d_in[1];   // N x 3
    const int*   grid_coord = (const int*)d_in[2];     // N x 3
    const int*   batch      = (const int*)d_in[3];     // N
    const float* W          = (const float*)d_in[4];   // 32 x 64
    const float* bias       = (const float*)d_in[5];   // 64

    const int N = in_sizes[0] / CIN;

    // output layout: feat_agg [N*64] | coord_agg [N*3] | cluster [N]
    float* outFeat    = (float*)d_out;
    float* outCoord   = outFeat + (size_t)N * COUT;
    float* outCluster = outCoord + (size_t)N * 3;

    // workspace layout (uint32 units)
    char* ws = (char*)d_ws;
    unsigned* bitmap    = (unsigned*)ws;                           // 128 MB
    unsigned* groupPref = bitmap + NWORDS;                         // 16 MB
    unsigned* blockSums = groupPref + NGROUPS;                     // 64 KB
    unsigned* keys      = blockSums + (NGROUPS / 256);             // 4 MB
    unsigned* cluster   = keys + N;                                // 4 MB
    unsigned* counts    = cluster + N;                             // 4 MB
    (void)ws_size; (void)n_in; (void)out_size;

    auto blocks = [](size_t n) { return (unsigned)((n + 255) / 256); };

    // init: clear bitmap + counts, coord sums, set feat region to ord(-inf)
    k_fill_u32<<<blocks(NWORDS), 256, 0, stream>>>(bitmap, 0u, NWORDS);
    k_fill_u32<<<blocks(N), 256, 0, stream>>>(counts, 0u, (size_t)N);
    k_fill_u32<<<blocks((size_t)N * COUT), 256, 0, stream>>>(
        (unsigned*)outFeat, ORD_NEG_INF, (size_t)N * COUT);
    k_fill_u32<<<blocks((size_t)N * 3), 256, 0, stream>>>(
        (unsigned*)outCoord, 0u, (size_t)N * 3);

    // voxel keys + bitmap population
    k_keys_bits<<<blocks(N), 256, 0, stream>>>(grid_coord, batch, keys, bitmap, N);

    // hierarchical popcount prefix over the bitmap
    k_group_sums<<<NGROUPS / 256, 256, 0, stream>>>(bitmap, groupPref);
    k_scan256<<<NGROUPS / 256, 256, 0, stream>>>(groupPref, blockSums, NGROUPS);
    k_scan_single<<<1, 256, 0, stream>>>(blockSums, NGROUPS / 256);
    k_add_offsets<<<NGROUPS / 256, 256, 0, stream>>>(groupPref, blockSums, NGROUPS);

    // cluster ranks, counts, coord sums
    k_cluster<<<blocks(N), 256, 0, stream>>>(keys, bitmap, groupPref, coord,
                                             cluster, counts, outCoord,
                                             outCluster, N);

    // fused WMMA GEMM + atomic segment-max  (N divisible by 16 -> 62500 waves)
    unsigned nWaves = (unsigned)((N + 15) / 16);
    unsigned gemmBlocks = (nWaves + 7) / 8;
    k_gemm_segmax<<<gemmBlocks, 256, 0, stream>>>(feat, W, bias, cluster,
                                                  (unsigned*)outFeat, N);

    // finalize
    k_finalize_feat<<<blocks((size_t)N * COUT), 256, 0, stream>>>(outFeat, counts, N);
    k_finalize_coord<<<blocks((size_t)N * 3), 256, 0, stream>>>(outCoord, counts, N);
}